// JResCOPAttn_90288802496809
// MI455X (gfx1250) — compile-verified
//
#include <hip/hip_runtime.h>
#include <hip/hip_bf16.h>

typedef __attribute__((ext_vector_type(2))) float v2f;
typedef __attribute__((ext_vector_type(8))) float v8f;

constexpr int Lr  = 1024;   // sequence length
constexpr int Dr  = 128;    // channels
constexpr int PAD = 132;    // LDS row stride (floats): even (b64 ok), 132%64=4 -> conflict-free
constexpr float EPSR = 1e-5f;

// ---------------------------------------------------------------------------
// Phase 0: a = x @ Wl^T + bl ;  tx = x @ Wl2^T + bl2     (both [L, D])
// One block per 16-row tile of x. 8 waves; wave w computes d-slice [16w,16w+16)
// for BOTH outputs (shared A operand).
// ---------------------------------------------------------------------------
__global__ void __launch_bounds__(256)
cop_phase0(const float* __restrict__ x,
           const float* __restrict__ Wl,  const float* __restrict__ bl,
           const float* __restrict__ Wl2, const float* __restrict__ bl2,
           float* __restrict__ a_ws, float* __restrict__ tx_ws) {
  __shared__ float xt[16 * PAD];

  const int t    = blockIdx.x;          // row tile
  const int tid  = threadIdx.x;         // 0..255
  const int wave = tid >> 5;
  const int lane = tid & 31;

  // cooperative load of x tile [16 x 128]
  for (int i = tid; i < 16 * Dr; i += 256) {
    const int r = i >> 7, k = i & 127;
    xt[r * PAD + k] = x[(t * 16 + r) * Dr + k];
  }
  __syncthreads();

  const int N     = lane & 15;          // B/C column, also A row (f32 16x16x4 layout)
  const int hi    = lane >> 4;          // K half: lanes 16..31 hold K+2,K+3
  const int dbase = wave * 16;
  const int d     = dbase + N;

  v8f acc1 = {};
  v8f acc2 = {};
#pragma unroll
  for (int k0 = 0; k0 < Dr; k0 += 4) {
    const int kk = k0 + 2 * hi;
    const v2f av = *(const v2f*)&xt[N * PAD + kk];          // A: row M=lane%16, K pair
    const v2f b1 = *(const v2f*)&Wl [d * Dr + kk];          // B: col N=d, K pair (Wl^T)
    const v2f b2 = *(const v2f*)&Wl2[d * Dr + kk];
    acc1 = __builtin_amdgcn_wmma_f32_16x16x4_f32(false, av, false, b1, (short)0, acc1, false, false);
    acc2 = __builtin_amdgcn_wmma_f32_16x16x4_f32(false, av, false, b2, (short)0, acc2, false, false);
  }

  const float bb1 = bl[d];
  const float bb2 = bl2[d];
#pragma unroll
  for (int v = 0; v < 8; ++v) {
    const int row = t * 16 + v + 8 * hi;                    // C layout: M = v + 8*(lane>=16)
    a_ws [row * Dr + d] = acc1[v] + bb1;
    tx_ws[row * Dr + d] = acc2[v] + bb2;
  }
}

// ---------------------------------------------------------------------------
// Phase 1: one block per output row c.
//   B operand breg[k] = Wlo[d,k] * a_c[k]  held entirely in VGPRs (invariant
//   across all 64 a-tiles -> loaded/built exactly once per wave).
//   for each a-tile (16 rows):  tm = Atile @ breg  via WMMA f32 16x16x4
//   acc[d] += (tm[a,d] + blo[d]) * mask[c,a] * tx[a,d]
//   y = x + acc ; LayerNorm(D) -> out
// ---------------------------------------------------------------------------
__global__ void __launch_bounds__(256)
cop_phase1(const float* __restrict__ x, const int* __restrict__ mask,
           const float* __restrict__ Wlo, const float* __restrict__ blo,
           const float* __restrict__ a_ws, const float* __restrict__ tx_ws,
           const float* __restrict__ gamma, const float* __restrict__ beta,
           float* __restrict__ out) {
  __shared__ float At[16 * PAD];        // 8.4 KB   a rows tile
  __shared__ float Tt[16 * PAD];        // 8.4 KB   tx rows tile
  __shared__ float maskf[Lr];           // 4 KB
  __shared__ float ac[Dr];              // a_c row (broadcast reads)
  __shared__ float ybuf[Dr];

  const int c    = blockIdx.x;
  const int tid  = threadIdx.x;
  const int wave = tid >> 5;
  const int lane = tid & 31;

  if (tid < Dr) ac[tid] = a_ws[c * Dr + tid];
  // mask row -> float 0/1
  for (int i = tid; i < Lr; i += 256)
    maskf[i] = (mask[c * Lr + i] != 0) ? 1.0f : 0.0f;
  __syncthreads();

  const int N  = lane & 15;
  const int hi = lane >> 4;
  const int d  = wave * 16 + N;
  const float bloN = blo[d];

  // Build this wave's B operand in registers: breg[s] covers K = 4s+2*hi .. +1
  v2f breg[32];
#pragma unroll
  for (int s = 0; s < 32; ++s) {
    const int kk = 4 * s + 2 * hi;
    const v2f w  = *(const v2f*)&Wlo[d * Dr + kk];   // Wlo row d (L2-resident)
    const v2f av = *(const v2f*)&ac[kk];             // LDS broadcast
    breg[s] = w * av;
  }

  float accN = 0.0f;                    // partial of sum_a over this lane's M-half

  for (int t = 0; t < Lr / 16; ++t) {
    __syncthreads();                    // all waves done reading previous tile
    for (int i = tid; i < 16 * Dr; i += 256) {
      const int r = i >> 7, k = i & 127;
      At[r * PAD + k] = a_ws [(t * 16 + r) * Dr + k];
      Tt[r * PAD + k] = tx_ws[(t * 16 + r) * Dr + k];
    }
    if (t + 1 < Lr / 16) {              // hint next tile into cache
      __builtin_prefetch(&a_ws [(t * 16 + 16) * Dr + (tid & 127)], 0, 1);
      __builtin_prefetch(&tx_ws[(t * 16 + 16) * Dr + (tid & 127)], 0, 1);
    }
    __syncthreads();

    v8f acc = {};
#pragma unroll
    for (int s = 0; s < 32; ++s) {
      const int kk = 4 * s + 2 * hi;
      const v2f av2 = *(const v2f*)&At[N * PAD + kk];     // A: M=lane%16
      acc = __builtin_amdgcn_wmma_f32_16x16x4_f32(false, av2, false, breg[s], (short)0, acc, false, false);
    }
    // fold (tm + blo)*mask*tx for the 8 a-rows this lane holds
#pragma unroll
    for (int v = 0; v < 8; ++v) {
      const int m    = v + 8 * hi;
      const int arow = t * 16 + m;
      accN += (acc[v] + bloN) * maskf[arow] * Tt[m * PAD + d];
    }
  }

  // combine the two M-halves: lanes l and l^16 hold the same d
  accN += __shfl_xor(accN, 16, 32);
  if (hi == 0) ybuf[d] = accN;
  __syncthreads();

  // residual + LayerNorm over D=128, done by wave 0 (4 d's per lane)
  if (wave == 0) {
    float y0[4];
    float s = 0.0f;
#pragma unroll
    for (int j = 0; j < 4; ++j) {
      const int dd = lane + 32 * j;
      y0[j] = x[c * Dr + dd] + ybuf[dd];
      s += y0[j];
    }
#pragma unroll
    for (int off = 16; off >= 1; off >>= 1) s += __shfl_xor(s, off, 32);
    const float mu = s * (1.0f / Dr);
    float vs = 0.0f;
#pragma unroll
    for (int j = 0; j < 4; ++j) { const float dl = y0[j] - mu; vs += dl * dl; }
#pragma unroll
    for (int off = 16; off >= 1; off >>= 1) vs += __shfl_xor(vs, off, 32);
    const float inv = rsqrtf(vs * (1.0f / Dr) + EPSR);
#pragma unroll
    for (int j = 0; j < 4; ++j) {
      const int dd = lane + 32 * j;
      out[c * Dr + dd] = (y0[j] - mu) * inv * gamma[dd] + beta[dd];
    }
  }
}

// ---------------------------------------------------------------------------
extern "C" void kernel_launch(void* const* d_in, const int* in_sizes, int n_in,
                              void* d_out, int out_size, void* d_ws, size_t ws_size,
                              hipStream_t stream) {
  const float* x     = (const float*)d_in[0];
  const int*   mask  = (const int*)  d_in[1];
  const float* Wl    = (const float*)d_in[2];
  const float* bl    = (const float*)d_in[3];
  const float* Wlo   = (const float*)d_in[4];
  const float* blo   = (const float*)d_in[5];
  const float* Wl2   = (const float*)d_in[6];
  const float* bl2   = (const float*)d_in[7];
  const float* gamma = (const float*)d_in[8];
  const float* beta  = (const float*)d_in[9];
  float* out = (float*)d_out;

  float* a_ws  = (float*)d_ws;                       // [L, D]
  float* tx_ws = a_ws + (size_t)Lr * Dr;             // [L, D]

  cop_phase0<<<Lr / 16, 256, 0, stream>>>(x, Wl, bl, Wl2, bl2, a_ws, tx_ws);
  cop_phase1<<<Lr, 256, 0, stream>>>(x, mask, Wlo, blo, a_ws, tx_ws, gamma, beta, out);
}